// ScatterReprTransform_83966610637148
// MI455X (gfx1250) — compile-verified
//
#include <hip/hip_runtime.h>
#include <stdint.h>

// Problem geometry (from reference): D = 128 features, groups of ~32 rows.
#define DIMS            128
#define WAVES_PER_BLOCK 8
#define STAGE_ROWS      8     // rows gathered per async double-buffer stage

// ---------------------------------------------------------------------------
// Kernel 1: exclusive prefix sum of per-group counts (general segment start
// offsets). Single block, sequential 1024-wide tiles with a running carry.
// G = 32768 -> 32 tiles; microseconds, off the critical path.
// ---------------------------------------------------------------------------
__global__ __launch_bounds__(1024) void scan_counts_kernel(
    const int* __restrict__ counts, int* __restrict__ offsets, int G_)
{
    __shared__ int tile[1024];
    __shared__ int carry;
    if (threadIdx.x == 0) carry = 0;
    __syncthreads();

    for (int base = 0; base < G_; base += 1024) {
        const int i = base + (int)threadIdx.x;
        const int v = (i < G_) ? counts[i] : 0;
        tile[threadIdx.x] = v;
        __syncthreads();
        const int c = carry;
        for (int off = 1; off < 1024; off <<= 1) {
            const int t = (threadIdx.x >= (unsigned)off) ? tile[threadIdx.x - off] : 0;
            __syncthreads();
            tile[threadIdx.x] += t;
            __syncthreads();
        }
        if (i < G_) offsets[i] = c + tile[threadIdx.x] - v;   // exclusive
        __syncthreads();
        if (threadIdx.x == 0) carry = c + tile[1023];
        __syncthreads();
    }
}

// ---------------------------------------------------------------------------
// Kernel 2: gather + segment-sum. One wave per group; each lane owns a
// float4 slice of the 128-wide row. The whole group's index list (<=32)
// is loaded with ONE coalesced b32 load (lane l -> ind[start+l]) and row
// indices are broadcast via v_readlane_b32 (uniform SGPR lane select), so
// the async-gather issue loop has no dependent memory loads in it.
// Rows are gathered 8-at-a-time into a double-buffered LDS stage with
// GLOBAL_LOAD_ASYNC_TO_LDS_B128 (one instruction moves a full 512B row:
// 32 lanes x 16B), overlapped with the previous stage's accumulation via
// s_wait_asynccnt partial waits.
// ---------------------------------------------------------------------------
__global__ __launch_bounds__(32 * WAVES_PER_BLOCK) void gather_segsum_async(
    const float* __restrict__ repr,
    const int*   __restrict__ ind,
    const int*   __restrict__ offsets,
    const int*   __restrict__ counts,
    float*       __restrict__ out,
    int G_)
{
    // 8 waves x 2 buffers x 8 rows x 128 f32 = 64 KB LDS per workgroup.
    __shared__ float buf[WAVES_PER_BLOCK][2][STAGE_ROWS][DIMS];

    const int wave = threadIdx.x >> 5;
    const int lane = threadIdx.x & 31;
    const int g    = blockIdx.x * WAVES_PER_BLOCK + wave;
    if (g >= G_) return;

    const int start = __builtin_amdgcn_readfirstlane(offsets[g]);
    const int count = __builtin_amdgcn_readfirstlane(counts[g]);

    float4 acc;
    acc.x = 0.f; acc.y = 0.f; acc.z = 0.f; acc.w = 0.f;

    const uint64_t reprBase = (uint64_t)(uintptr_t)repr;
    const uint32_t laneByte = (uint32_t)lane * 16u;

    if (count <= 32) {
        // ---- fast path (harness case: count == 32) ----
        // One coalesced load grabs the whole group's index list.
        int myidx = 0;
        if (lane < count) myidx = ind[start + lane];

        // gfx1250 flat LDS addresses carry the LDS byte offset in addr[31:0].
        const uint32_t lds0 = (uint32_t)(uintptr_t)(&buf[wave][0][0][0]) + laneByte;
        const uint32_t stageStride = (uint32_t)(STAGE_ROWS * DIMS * sizeof(float));

        const int full = count >> 3;   // number of 8-row stages

        // ---- prime stage 0 into buffer 0 ----
        if (full > 0) {
            #pragma unroll
            for (int r = 0; r < STAGE_ROWS; ++r) {
                const int row = __builtin_amdgcn_readlane(myidx, r);
                const uint64_t ga = reprBase + (uint64_t)(uint32_t)row * (DIMS * 4u)
                                  + (uint64_t)laneByte;
                const uint32_t la = lds0 + (uint32_t)r * (DIMS * (uint32_t)sizeof(float));
                asm volatile("global_load_async_to_lds_b128 %0, %1, off"
                             :: "v"(la), "v"(ga) : "memory");
            }
        }

        for (int s = 0; s < full; ++s) {
            const int cb = s & 1;
            const int nb = (s + 1) & 1;

            if (s + 1 < full) {
                // Issue next stage while current stage's gathers are in flight.
                const int base = (s + 1) * STAGE_ROWS;
                #pragma unroll
                for (int r = 0; r < STAGE_ROWS; ++r) {
                    const int row = __builtin_amdgcn_readlane(myidx, base + r);
                    const uint64_t ga = reprBase + (uint64_t)(uint32_t)row * (DIMS * 4u)
                                      + (uint64_t)laneByte;
                    const uint32_t la = lds0 + (uint32_t)nb * stageStride +
                                        (uint32_t)r * (DIMS * (uint32_t)sizeof(float));
                    asm volatile("global_load_async_to_lds_b128 %0, %1, off"
                                 :: "v"(la), "v"(ga) : "memory");
                }
                // Async loads complete in order: <=8 outstanding => stage s landed.
                asm volatile("s_wait_asynccnt 8" ::: "memory");
            } else {
                asm volatile("s_wait_asynccnt 0" ::: "memory");
            }

            // Accumulate current stage from LDS: ds_load_b128 + 4 adds per row.
            #pragma unroll
            for (int r = 0; r < STAGE_ROWS; ++r) {
                const float4 v = *(const float4*)(&buf[wave][cb][r][lane * 4]);
                acc.x += v.x; acc.y += v.y; acc.z += v.z; acc.w += v.w;
            }
        }

        // Remainder rows (count % 8 != 0): direct gathered b128 loads.
        for (int r = full * STAGE_ROWS; r < count; ++r) {
            const int row = __builtin_amdgcn_readlane(myidx, r);
            const float4 v = *((const float4*)(repr + (size_t)(uint32_t)row * DIMS) + lane);
            acc.x += v.x; acc.y += v.y; acc.z += v.z; acc.w += v.w;
        }
    } else {
        // ---- generic path (count > 32; not hit by this harness) ----
        for (int r = 0; r < count; ++r) {
            const int row = __builtin_amdgcn_readfirstlane(ind[start + r]);
            const float4 v = *((const float4*)(repr + (size_t)(uint32_t)row * DIMS) + lane);
            acc.x += v.x; acc.y += v.y; acc.z += v.z; acc.w += v.w;
        }
    }

    // Coalesced 512B row store (global_store_b128 per lane).
    *((float4*)(out + (size_t)g * DIMS) + lane) = acc;
}

// ---------------------------------------------------------------------------
// Launch
//   d_in[0] = repr      (N*D f32)
//   d_in[1] = ind       (E int)
//   d_in[2] = ind_block (G int)
//   d_out   = out       (G*D f32)
//   d_ws    = G int exclusive offsets
// ---------------------------------------------------------------------------
extern "C" void kernel_launch(void* const* d_in, const int* in_sizes, int n_in,
                              void* d_out, int out_size, void* d_ws, size_t ws_size,
                              hipStream_t stream)
{
    const float* repr      = (const float*)d_in[0];
    const int*   ind       = (const int*)d_in[1];
    const int*   ind_block = (const int*)d_in[2];
    float*       out       = (float*)d_out;
    const int    G_        = in_sizes[2];
    int*         offsets   = (int*)d_ws;

    scan_counts_kernel<<<1, 1024, 0, stream>>>(ind_block, offsets, G_);

    const int blocks = (G_ + WAVES_PER_BLOCK - 1) / WAVES_PER_BLOCK;
    gather_segsum_async<<<blocks, 32 * WAVES_PER_BLOCK, 0, stream>>>(
        repr, ind, offsets, ind_block, out, G_);
}